// ButterflyConv2dBBT_28853590295183
// MI455X (gfx1250) — compile-verified
//
#include <hip/hip_runtime.h>

// Shapes from the reference
#define BB   16
#define CIN  256
#define HH   32
#define WW   32
#define OCN  256
#define KPOS 9            // 3x3 patch positions
#define MST  8            // butterfly stages per factor
#define KDIM (KPOS * CIN) // 2304 GEMM-K
#define KCHUNK 16
#define NCHUNKS (KDIM / KCHUNK) // 144
#define MTILE 64          // pixels per block (two image rows)

typedef __attribute__((ext_vector_type(2))) float v2f;
typedef __attribute__((ext_vector_type(4))) float v4f;
typedef __attribute__((ext_vector_type(8))) float v8f;

// ---------------------------------------------------------------------------
// Phase 1: compose the two untied butterfly factors into dense per-position
// weights, stored K-quad packed for the WMMA B-fragment layout:
//   Wdp[(k>>2)*1024 + oc*4 + (k&3)] = (1/9) * (B2_s(B1_s(e_cin)))[oc]
// with k = s*256 + cin. One block per (s, cin); 128 threads = 128 pairs.
// ---------------------------------------------------------------------------
__global__ __launch_bounds__(128) void bfly_build_weights(
    const float* __restrict__ tw1, const float* __restrict__ tw2,
    float* __restrict__ Wdp) {
  const int s = blockIdx.x >> 8;
  const int c = blockIdx.x & 255;
  __shared__ float v[256];
  const int p = threadIdx.x; // pair index 0..127
  v[2 * p]     = (2 * p     == c) ? 1.f : 0.f;
  v[2 * p + 1] = (2 * p + 1 == c) ? 1.f : 0.f;
  __syncthreads();
  #pragma unroll
  for (int st = 0; st < 2 * MST; ++st) {
    // factor 1: decreasing stride (ls = 7..0); factor 2: increasing (0..7)
    const int ls = (st < MST) ? (MST - 1 - st) : (st - MST);
    const float* tw = (st < MST) ? tw1 : tw2;
    const float* T = tw + (((s * MST + ls) * 128) + p) * 4;
    const int stride = 1 << ls;
    const int gi = p >> ls;
    const int k  = p & (stride - 1);
    const int i0 = (gi << (ls + 1)) + k;
    const int i1 = i0 + stride;
    const float a0 = v[i0], a1 = v[i1];
    const float t00 = T[0], t01 = T[1], t10 = T[2], t11 = T[3];
    v[i0] = t00 * a0 + t01 * a1;
    v[i1] = t10 * a0 + t11 * a1;
    __syncthreads();
  }
  const float sc = 1.f / 9.f;
  const int k = s * 256 + c;
  float* dst = Wdp + ((size_t)(k >> 2) * (OCN * 4)) + (k & 3);
  dst[(2 * p) * 4]     = v[2 * p]     * sc;
  dst[(2 * p + 1) * 4] = v[2 * p + 1] * sc;
}

// bias_mean[oc] = (1/9) * sum_s bias2[s, oc]
__global__ void bfly_bias_mean(const float* __restrict__ bias2,
                               float* __restrict__ bm) {
  const int oc = threadIdx.x;
  float a = 0.f;
  #pragma unroll
  for (int s = 0; s < KPOS; ++s) a += bias2[s * OCN + oc];
  bm[oc] = a * (1.f / 9.f);
}

// ---------------------------------------------------------------------------
// Phase 2: implicit-GEMM 3x3 conv via V_WMMA_F32_16X16X4_F32.
// Block = 256 threads (8 waves). Tile: 64 pixels (two image rows) x 256 oc.
// Wave(w): rows [ (w&3)*16, +16 ), cols [ (w>>2)*128, +128 ) -> 8 C-tiles.
// Per K-chunk per wave: 4 a-frag b64 + 32 b-frag b64 loads feed 32 WMMAs.
// ---------------------------------------------------------------------------
__global__ __launch_bounds__(256) void bfly_conv_wmma(
    const float* __restrict__ in, const float* __restrict__ Wdp,
    const float* __restrict__ bm, float* __restrict__ out) {
  union Sm {
    struct {
      float A[2][MTILE][18]; // [buf][pixel][k]; stride 18: 8B-aligned pairs,
                             // conflict-free for both store and frag read
      float B[2][4096];      // [buf][k4*1024 + oc*4 + kq] (K-quad packed)
    } ab;
    float C[128][67];        // epilogue transpose (one 128-oc half at a time)
  };
  __shared__ Sm sm;

  const int t    = threadIdx.x;
  const int lane = t & 31;
  const int wave = t >> 5;
  const int mBase = (wave & 3) << 4;  // 0/16/32/48
  const int nOff  = (wave >> 2) << 7; // 0 / 128

  const int p0   = blockIdx.x << 6;   // first pixel of this block
  const int bImg = p0 >> 10;
  const int y0   = (p0 >> 5) & 31;    // covers rows y0, y0+1 (y0 even)

  // loader assignments
  const int mA = t & 63;        // pixel index within tile
  const int kA = t >> 6;        // k row 0..3 (+4q passes)
  const int xA = mA & 31;
  const int yAofs = mA >> 5;    // 0/1: which image row

  v8f acc[8];
  #pragma unroll
  for (int i = 0; i < 8; ++i) acc[i] = (v8f){};

  auto fetchA = [&](int ch, float* r) {
    const int k0 = ch * KCHUNK;
    const int s  = k0 >> 8;     // patch position (uniform per chunk)
    const int c0 = k0 & 255;
    const int dy = s / 3 - 1, dx = s % 3 - 1;
    const int yy = y0 + yAofs + dy;
    const int xx = xA + dx;
    const bool ok = ((unsigned)xx < 32u) && ((unsigned)yy < 32u);
    #pragma unroll
    for (int q = 0; q < 4; ++q) {
      const int c = c0 + kA + 4 * q;
      r[q] = ok ? in[((bImg * CIN + c) * HH + yy) * WW + xx] : 0.f;
    }
  };
  auto fetchB = [&](int ch, v4f* r) {
    const float* src = Wdp + (size_t)ch * 4096 + t * 4; // packed, coalesced
    #pragma unroll
    for (int q = 0; q < 4; ++q) r[q] = *(const v4f*)(src + q * 1024);
  };
  auto stageAB = [&](int buf, const float* rA, const v4f* rB) {
    #pragma unroll
    for (int q = 0; q < 4; ++q) sm.ab.A[buf][mA][kA + 4 * q] = rA[q];
    float* dst = sm.ab.B[buf] + t * 4;
    #pragma unroll
    for (int q = 0; q < 4; ++q) *(v4f*)(dst + q * 1024) = rB[q];
  };
  auto compute = [&](int buf) {
    const int koff = (lane >> 4) << 1; // lanes 0-15 -> K0/K1, 16-31 -> K2/K3
    const int mRow = mBase + (lane & 15);
    const int colL = lane & 15;
    #pragma unroll
    for (int kk4 = 0; kk4 < KCHUNK; kk4 += 4) {
      v2f a = *(const v2f*)&sm.ab.A[buf][mRow][kk4 + koff]; // b64, 8B aligned
      const float* bbase =
          sm.ab.B[buf] + ((kk4 >> 2) * 256 + nOff + colL) * 4 + koff;
      #pragma unroll
      for (int nt = 0; nt < 8; ++nt) {
        v2f b = *(const v2f*)(bbase + nt * 64); // b64, 8B aligned
        acc[nt] = __builtin_amdgcn_wmma_f32_16x16x4_f32(
            false, a, false, b, (short)0, acc[nt], false, false);
      }
    }
  };

  // prologue: chunk 0 into buffer 0
  {
    float rA[4]; v4f rB[4];
    fetchA(0, rA);
    fetchB(0, rB);
    stageAB(0, rA, rB);
    __syncthreads();
  }
  // main loop: prefetch chunk ch+1 to regs, compute chunk ch, stage, barrier
  for (int ch = 0; ch < NCHUNKS; ++ch) {
    const int cur = ch & 1;
    const bool more = (ch + 1 < NCHUNKS);
    float rA[4] = {}; v4f rB[4] = {};
    if (more) { fetchA(ch + 1, rA); fetchB(ch + 1, rB); }
    compute(cur);
    if (more) stageAB(cur ^ 1, rA, rB);
    __syncthreads();
  }

  // epilogue: transpose through LDS in two 128-oc halves (reuses A/B LDS),
  // then fully-coalesced float4 stores.
  const int xmBase = mBase + ((lane >> 4) << 3); // lanes>=16 hold rows r+8
  const int colL   = lane & 15;
  #pragma unroll
  for (int half = 0; half < 2; ++half) {
    if ((wave >> 2) == half) {
      #pragma unroll
      for (int nt = 0; nt < 8; ++nt) {
        const int cc = nt * 16 + colL; // 0..127 within this half
        #pragma unroll
        for (int r = 0; r < 8; ++r) sm.C[cc][xmBase + r] = acc[nt][r];
      }
    }
    __syncthreads();
    const int ocBase = half << 7;
    #pragma unroll
    for (int pass = 0; pass < 8; ++pass) {
      const int occ = pass * 16 + (t >> 4); // 0..127
      const int xq  = (t & 15) << 2;        // 0..60, stays within one row of 32
      const int oc  = ocBase + occ;
      const float bv = bm[oc];
      v4f vv;
      vv.x = sm.C[occ][xq]     + bv;
      vv.y = sm.C[occ][xq + 1] + bv;
      vv.z = sm.C[occ][xq + 2] + bv;
      vv.w = sm.C[occ][xq + 3] + bv;
      *(v4f*)&out[((bImg * OCN + oc) * HH + (y0 + (xq >> 5))) * WW + (xq & 31)] = vv;
    }
    __syncthreads();
  }
}

// ---------------------------------------------------------------------------
// Launch: build packed dense weights + mean bias in workspace, then WMMA conv.
// Workspace need: (2304*256 + 256) floats ~= 2.36 MB.
// ---------------------------------------------------------------------------
extern "C" void kernel_launch(void* const* d_in, const int* in_sizes, int n_in,
                              void* d_out, int out_size, void* d_ws, size_t ws_size,
                              hipStream_t stream) {
  const float* in  = (const float*)d_in[0];
  const float* tw1 = (const float*)d_in[1];
  const float* tw2 = (const float*)d_in[2];
  const float* b2  = (const float*)d_in[3];
  float* outp = (float*)d_out;
  float* Wdp = (float*)d_ws;                 // packed [576][256][4]
  float* bm  = Wdp + (size_t)KDIM * OCN;     // [256]

  bfly_build_weights<<<KPOS * CIN, 128, 0, stream>>>(tw1, tw2, Wdp);
  bfly_bias_mean<<<1, OCN, 0, stream>>>(b2, bm);
  bfly_conv_wmma<<<(BB * HH * WW) / MTILE, 256, 0, stream>>>(in, Wdp, bm, outp);
}